// ComplExDecoder_15040975470742
// MI455X (gfx1250) — compile-verified
//
#include <hip/hip_runtime.h>

typedef __attribute__((ext_vector_type(2))) float v2f;
typedef __attribute__((ext_vector_type(4))) float v4f;
typedef __attribute__((ext_vector_type(8))) float v8f;
typedef __attribute__((ext_vector_type(4))) int   v4i;

typedef __attribute__((address_space(1))) v4i g_v4i;   // global int4
typedef __attribute__((address_space(3))) v4i l_v4i;   // LDS int4

#define DFEAT 64
#define NREL  100
#define RELF  (NREL * DFEAT)      // 6400 floats per table
#define RELV4 (RELF / 4)          // 1600 float4 per table

#if __has_builtin(__builtin_amdgcn_global_load_async_to_lds_b128) && \
    __has_builtin(__builtin_amdgcn_s_wait_asynccnt)
#define HAVE_ASYNC_LDS 1
#else
#define HAVE_ASYNC_LDS 0
#endif

__global__ __launch_bounds__(256) void complex_score_wmma(
    const float* __restrict__ x_re, const float* __restrict__ x_im,
    const float* __restrict__ R_re, const float* __restrict__ R_im,
    const int*   __restrict__ ei,   const int*   __restrict__ et,
    float* __restrict__ out, int E)
{
    // Stage relation tables in LDS: 2 * 25.6 KB = 51.2 KB (<< 320 KB/WGP)
    __shared__ __align__(16) float sRre[RELF];
    __shared__ __align__(16) float sRim[RELF];

    const int tid = threadIdx.x;
#if HAVE_ASYNC_LDS
    {
        // CDNA5 async copy engine: GLOBAL_LOAD_ASYNC_TO_LDS_B128, no VGPR
        // round-trip, tracked by ASYNCcnt (per-wave), then workgroup barrier.
        for (int i = tid; i < RELV4; i += 256) {
            __builtin_amdgcn_global_load_async_to_lds_b128(
                (g_v4i*)(R_re + 4 * i), (l_v4i*)(sRre + 4 * i),
                /*offset=*/0, /*cpol=*/0);
            __builtin_amdgcn_global_load_async_to_lds_b128(
                (g_v4i*)(R_im + 4 * i), (l_v4i*)(sRim + 4 * i),
                /*offset=*/0, /*cpol=*/0);
        }
        __builtin_amdgcn_s_wait_asynccnt(0);   // this wave's copies complete
    }
#else
    {
        const v4f* g0 = (const v4f*)R_re;
        const v4f* g1 = (const v4f*)R_im;
        v4f* l0 = (v4f*)sRre;
        v4f* l1 = (v4f*)sRim;
        for (int i = tid; i < RELV4; i += 256) { l0[i] = g0[i]; l1[i] = g1[i]; }
    }
#endif
    __syncthreads();

    const int lane = tid & 31;          // wave32
    const int wave = tid >> 5;          // 8 waves / block
    const int m    = lane & 15;         // edge within group of 16
    const int h    = lane >> 4;         // which half of the 64-dim row

    const int group = blockIdx.x * 8 + wave;
    const int e0    = group * 16;
    const int em    = e0 + m;
    const int ec    = (em < E) ? em : (E - 1);   // clamp for safe gathers

    const int src = ei[ec];
    const int dst = ei[E + ec];
    const int rel = et[ec];

    const v4f* sre4 = (const v4f*)(x_re + (size_t)src * DFEAT + 32 * h);
    const v4f* sim4 = (const v4f*)(x_im + (size_t)src * DFEAT + 32 * h);
    const v4f* ore4 = (const v4f*)(x_re + (size_t)dst * DFEAT + 32 * h);
    const v4f* oim4 = (const v4f*)(x_im + (size_t)dst * DFEAT + 32 * h);
    const v4f* rre4 = (const v4f*)(sRre + rel * DFEAT + 32 * h);
    const v4f* rim4 = (const v4f*)(sRim + rel * DFEAT + 32 * h);

    // Two accumulators per lane -> A-matrix layout of V_WMMA_F32_16X16X4_F32:
    // lane m   holds A[m][0], A[m][1]; lane m+16 holds A[m][2], A[m][3].
    float a0 = 0.0f, a1 = 0.0f;
#pragma unroll
    for (int t = 0; t < 8; ++t) {
        v4f sre = sre4[t], sim = sim4[t];
        v4f ore = ore4[t], oim = oim4[t];
        v4f rre = rre4[t], rim = rim4[t];
#pragma unroll
        for (int j = 0; j < 4; ++j) {
            // sr_re = s_re*r_re - s_im*r_im ;  -sr_im = -(s_re*r_im + s_im*r_re)
            float srre  = fmaf(sre[j],  rre[j], -(sim[j] * rim[j]));
            float nsrim = fmaf(-sre[j], rim[j], -(sim[j] * rre[j]));
            a0 = fmaf(srre,  ore[j], a0);   // + sr_re*o_re
            a1 = fmaf(nsrim, oim[j], a1);   // - sr_im*o_im
        }
    }

    // Reduce the 4 per-edge partials (2 lanes x 2 VGPRs) with a single
    // f32 WMMA against an all-ones B: D[m][n] = sum_k A[m][k].
    v2f A; A[0] = a0;   A[1] = a1;
    v2f B; B[0] = 1.0f; B[1] = 1.0f;
    v8f C = {};
    C = __builtin_amdgcn_wmma_f32_16x16x4_f32(
            /*neg_a=*/false, A, /*neg_b=*/false, B,
            /*c_mod=*/(short)0, C, /*reuse_a=*/false, /*reuse_b=*/false);

    // D layout: VGPR v, lanes 0-15 -> row v; lanes 16-31 -> row v+8.
    // Rows are replicated across columns, so column 0..7 of each half writes.
    const int s = lane & 7;
    float r01 = (s & 1) ? C[1] : C[0];
    float r23 = (s & 1) ? C[3] : C[2];
    float r45 = (s & 1) ? C[5] : C[4];
    float r67 = (s & 1) ? C[7] : C[6];
    float rlo = (s & 2) ? r23 : r01;
    float rhi = (s & 2) ? r67 : r45;
    float rv  = (s & 4) ? rhi : rlo;

    const int eout = e0 + s + h * 8;
    if (((lane & 15) < 8) && (eout < E)) out[eout] = rv;
}

extern "C" void kernel_launch(void* const* d_in, const int* in_sizes, int n_in,
                              void* d_out, int out_size, void* d_ws, size_t ws_size,
                              hipStream_t stream)
{
    const float* x_re = (const float*)d_in[0];
    const float* x_im = (const float*)d_in[1];
    const float* R_re = (const float*)d_in[2];
    const float* R_im = (const float*)d_in[3];
    const int*   ei   = (const int*)d_in[4];    // [2, E] flat: row0=src, row1=dst
    const int*   et   = (const int*)d_in[5];    // [E]
    float*       out  = (float*)d_out;

    const int E = in_sizes[5];
    const int groups = (E + 15) / 16;           // 16 edges per wave
    const int blocks = (groups + 7) / 8;        // 8 waves (256 thr) per block

    complex_score_wmma<<<blocks, 256, 0, stream>>>(x_re, x_im, R_re, R_im, ei, et, out, E);
}